// LearnableDecomposition_35768487641205
// MI455X (gfx1250) — compile-verified
//
#include <hip/hip_runtime.h>

typedef float v2f __attribute__((ext_vector_type(2)));
typedef float v8f __attribute__((ext_vector_type(8)));

#define Bn 32
#define Ln 2048
#define Cn 512
#define Kn 25
#define TILE 16
#define STRIP_TILES 8                          // L-tiles per wave strip
#define NSTRIPS (Ln / (TILE * STRIP_TILES))    // 16
#define CHUNKS 10                              // 40 input rows / 4 per WMMA K-step

// out[b, l, c] = sum_k w[k] * x[b, clamp(l + k - 12), c] + bias
// Banded-matmul via V_WMMA_F32_16X16X4_F32:
//   out_tile(16x16) = A(16x40) @ X(40x16),  A[i,j] = (0<=j-i<25) ? w[j-i] : 0
//
// Fragment slots are indexed modularly: slot s holds absolute chunk Q with
// Q % 10 == s. Live chunks at tile t are Q = 4t..4t+9 (10 consecutive), so
// the mapping is a bijection and, fully unrolled, needs zero register moves.
template <bool INTERIOR>
__device__ __forceinline__ void run_strip(
    const float* __restrict__ x_lane,  // x + batch*L*C + cbase (per-lane column)
    float* __restrict__ o_lane,        // out + batch*L*C + cbase + hiH*8*C
    const v2f (&a)[CHUNKS],
    int l_start, int off, float bval)
{
  // Interior strips: all rows l_start-12 .. l_start+139 are in-bounds, so every
  // access is base-pointer + compile-time immediate offset (fits IOFFSET i24).
  const float* xb_ = x_lane + (long)(l_start - 12 + off) * Cn;
  float*       ob_ = o_lane + (long)l_start * Cn;

  auto ld = [&](int Q, int sub) -> float {
    if (INTERIOR) {
      return xb_[(long)(4 * Q + sub) * Cn];          // constant index -> IOFFSET
    } else {
      int j = l_start - 12 + 4 * Q + off + sub;      // replicate pad (clamp)
      j = j < 0 ? 0 : (j > Ln - 1 ? Ln - 1 : j);
      return x_lane[(long)j * Cn];
    }
  };

  v2f bf[CHUNKS];
#pragma unroll
  for (int Q = 0; Q < CHUNKS; ++Q) {                 // initial window: Q = 0..9
    bf[Q % CHUNKS].x = ld(Q, 0);
    bf[Q % CHUNKS].y = ld(Q, 1);
  }

#pragma unroll
  for (int t = 0; t < STRIP_TILES; ++t) {
    v8f c = {};
#pragma unroll
    for (int q = 0; q < CHUNKS; ++q) {
      // 8 args: (neg_a, A, neg_b, B, c_mod, C, reuse_a, reuse_b)
      c = __builtin_amdgcn_wmma_f32_16x16x4_f32(
              false, a[q], false, bf[(4 * t + q) % CHUNKS], (short)0, c,
              false, false);
    }

    // Store 16x16 tile (+bias); C/D layout: VGPR r -> row r (+8 folded into ob_)
#pragma unroll
    for (int r = 0; r < 8; ++r) {
      __builtin_nontemporal_store(c[r] + bval,
                                  &ob_[(long)(t * TILE + r) * Cn]);
    }

    // Slide window by 16 rows: overwrite the 4 dead slots with Q = 4t+10..4t+13
    if (t != STRIP_TILES - 1) {
#pragma unroll
      for (int Q = 4 * t + 10; Q < 4 * t + 14; ++Q) {
        bf[Q % CHUNKS].x = ld(Q, 0);
        bf[Q % CHUNKS].y = ld(Q, 1);
      }
    }
  }
}

__global__ __launch_bounds__(256) void dwconv25_wmma_f32(
    const float* __restrict__ x,    // [32, 2048, 512]
    const float* __restrict__ w,    // [25]
    const float* __restrict__ bias, // [1]
    float* __restrict__ out)        // [32, 2048, 512]
{
  const int wave = blockIdx.x * (blockDim.x >> 5) + (threadIdx.x >> 5);
  const int lane = threadIdx.x & 31;
  const int n    = lane & 15;       // column (M for A-frag, N for B/C-frags)
  const int hiH  = lane >> 4;       // lane-half select
  const int off  = hiH << 1;        // K offset within a 4-chunk (0 or 2)

  const int ctile  = wave & 31;         // 32 channel tiles of 16
  const int lstrip = (wave >> 5) & 15;  // 16 strips of 128 L positions
  const int batch  = wave >> 9;         // 32 batches

  const float bval = bias[0];

  // A fragments (constant band matrix), 16x4 fp32 layout:
  // lanes 0-15: M=lane, v0=K0, v1=K1 ; lanes 16-31: M=lane-16, v0=K2, v1=K3
  v2f a[CHUNKS];
#pragma unroll
  for (int q = 0; q < CHUNKS; ++q) {
    int d0 = 4 * q + off - n;           // j - i
    int d1 = d0 + 1;
    a[q].x = (d0 >= 0 && d0 < Kn) ? w[d0] : 0.0f;
    a[q].y = (d1 >= 0 && d1 < Kn) ? w[d1] : 0.0f;
  }

  const long xb      = (long)batch * Ln * Cn;
  const long cbase   = (long)ctile * 16 + n;
  const int  l_start = lstrip * (TILE * STRIP_TILES);

  const float* x_lane = x + xb + cbase;
  float*       o_lane = out + xb + cbase + (long)hiH * 8 * Cn;

  if (lstrip != 0 && lstrip != NSTRIPS - 1) {
    run_strip<true >(x_lane, o_lane, a, l_start, off, bval);  // fast path
  } else {
    run_strip<false>(x_lane, o_lane, a, l_start, off, bval);  // clamped edges
  }
}

extern "C" void kernel_launch(void* const* d_in, const int* in_sizes, int n_in,
                              void* d_out, int out_size, void* d_ws, size_t ws_size,
                              hipStream_t stream) {
  (void)in_sizes; (void)n_in; (void)out_size; (void)d_ws; (void)ws_size;
  const float* x = (const float*)d_in[0];
  const float* w = (const float*)d_in[1];
  const float* b = (const float*)d_in[2];
  float*       o = (float*)d_out;

  // waves = 32 batches * 16 L-strips * 32 C-tiles = 16384; 8 waves / block
  const int total_waves = Bn * NSTRIPS * (Cn / 16);
  dim3 block(256);
  dim3 grid(total_waves / 8);
  hipLaunchKernelGGL(dwconv25_wmma_f32, grid, block, 0, stream, x, w, b, o);
}